// chamfer_3DDist_82935818486394
// MI455X (gfx1250) — compile-verified
//
#include <hip/hip_runtime.h>

typedef __attribute__((ext_vector_type(2))) float v2f;
typedef __attribute__((ext_vector_type(8))) float v8f;

#define NPTS 8192
#define NB   4
#define QT   2                 // 16-col query tiles resident per wave
#define NCHUNK 4               // split of the target dimension
#define WAVES_PER_BLOCK 8
#define THREADS (WAVES_PER_BLOCK * 32)
#define Q_PER_WAVE (16 * QT)   // 32 queries per wave

// Pack (x,y,z) into both roles:
//   pkA[i] = (-2x, -2y, -2z, |p|^2)   (streamed A-matrix rows, pre-scaled)
//   pkB[i] = ( x,   y,   z,  |p|^2)   (resident B columns + C norms)
__global__ void chamfer_pack_kernel(const float* __restrict__ xyz,
                                    float4* __restrict__ pkA,
                                    float4* __restrict__ pkB, int n) {
    int i = blockIdx.x * blockDim.x + threadIdx.x;
    if (i < n) {
        float x = xyz[3 * i + 0];
        float y = xyz[3 * i + 1];
        float z = xyz[3 * i + 2];
        float nn = x * x + y * y + z * z;
        pkA[i] = make_float4(-2.0f * x, -2.0f * y, -2.0f * z, nn);
        pkB[i] = make_float4(x, y, z, nn);
    }
}

// Partial pass. Queries live in the (loop-invariant) B matrix / C accumulator:
//   B column n = (x,y,z,1) of query n, C[:][n] = |q_n|^2 (broadcast down rows).
// Targets stream through A: row m = (-2x,-2y,-2z,|t_m|^2).
//   D[m][n] = |t_m|^2 + |q_n|^2 - 2 t.q  == squared distance, 256 pairs / WMMA.
// blockIdx.z encodes (direction, target-chunk); partial (min,argmin) -> workspace.
__global__ void __launch_bounds__(THREADS)
chamfer_partial_kernel(const float4* __restrict__ pkA1,
                       const float4* __restrict__ pkB1,
                       const float4* __restrict__ pkA2,
                       const float4* __restrict__ pkB2,
                       float* __restrict__ pval,
                       int* __restrict__ pidx,
                       int npts) {
    const int lane = threadIdx.x & 31;
    const int wave = threadIdx.x >> 5;
    const int hi   = lane >> 4;     // K-half for A/B operands; row-half (M +8) for D
    const int ln   = lane & 15;     // query column owned by this lane
    const int b    = blockIdx.y;
    const int dir  = blockIdx.z / NCHUNK;
    const int chk  = blockIdx.z % NCHUNK;

    // dir 0: queries = set1, targets = set2 ; dir 1: swapped.
    const float4* __restrict__ Qb = (dir ? pkB2 : pkB1) + (size_t)b * npts;
    const float*  __restrict__ Ta = (const float*)((dir ? pkA1 : pkA2) + (size_t)b * npts);

    const int qbase  = (blockIdx.x * WAVES_PER_BLOCK + wave) * Q_PER_WAVE;
    const int tchunk = npts / NCHUNK;
    const int tstart = chk * tchunk;
    const int tend   = tstart + tchunk;

    // Resident B columns and C norms (hoisted out of the target loop).
    v2f Bm[QT];
    v8f Cm[QT];
#pragma unroll
    for (int t = 0; t < QT; ++t) {
        float4 q = Qb[qbase + t * 16 + ln];
        Bm[t].x = hi ? q.z : q.x;
        Bm[t].y = hi ? 1.0f : q.y;
        float cc = q.w;
        Cm[t] = (v8f){ cc, cc, cc, cc, cc, cc, cc, cc };
    }

    float best[QT];
    int   bidx[QT];
#pragma unroll
    for (int t = 0; t < QT; ++t) { best[t] = __FLT_MAX__; bidx[t] = tstart; }

#pragma unroll 2
    for (int tb = tstart; tb < tend; tb += 16) {
        // A tile: one 8-byte load per lane (K0,K1 for lanes 0-15; K2,K3 for 16-31).
        const float2 av = *(const float2*)(Ta + 4 * (tb + ln) + 2 * hi);
        v2f Av; Av.x = av.x; Av.y = av.y;

#pragma unroll
        for (int t = 0; t < QT; ++t) {
            v8f D = __builtin_amdgcn_wmma_f32_16x16x4_f32(
                false, Av, false, Bm[t], (short)0, Cm[t], false, false);
#pragma unroll
            for (int r = 0; r < 8; ++r) {
                // slot r = target (tb + r + 8*hi) vs this lane's query column
                int  cand = tb + r + 8 * hi;
                float d   = D[r];
                bool lt   = d < best[t];   // raw dsq; ascending index + strict < = first occurrence
                best[t] = lt ? d : best[t];
                bidx[t] = lt ? cand : bidx[t];
            }
        }
    }

    // Merge the two row-halves: lanes ln and ln+16 hold the same query,
    // covering targets m%16<8 and m%16>=8. Tie -> smaller global index.
#pragma unroll
    for (int t = 0; t < QT; ++t) {
        float ov = __shfl_xor(best[t], 16, 32);
        int   oi = __shfl_xor(bidx[t], 16, 32);
        bool take = (ov < best[t]) || (ov == best[t] && oi < bidx[t]);
        best[t] = take ? ov : best[t];
        bidx[t] = take ? oi : bidx[t];
    }

    // Coalesced partial stores from lanes 0-15.
    if (lane < 16) {
        const size_t pbase = (((size_t)dir * NB + b) * NCHUNK + chk) * npts;
#pragma unroll
        for (int t = 0; t < QT; ++t) {
            pval[pbase + qbase + t * 16 + ln] = best[t];
            pidx[pbase + qbase + t * 16 + ln] = bidx[t];
        }
    }
}

// Combine NCHUNK partials per query (ascending chunk => first-occurrence ties),
// clamp, sqrt, emit.
__global__ void chamfer_reduce_kernel(const float* __restrict__ pval,
                                      const int* __restrict__ pidx,
                                      float* __restrict__ out, int npts) {
    const int total = 2 * NB * npts;
    int tid = blockIdx.x * blockDim.x + threadIdx.x;
    if (tid >= total) return;

    const int dir = tid / (NB * npts);
    const int rem = tid % (NB * npts);          // b * npts + m
    const size_t base = ((size_t)dir * NB * NCHUNK) * npts
                      + (size_t)(rem / npts) * NCHUNK * npts + (rem % npts);

    float v  = pval[base];
    int   ix = pidx[base];
#pragma unroll
    for (int c = 1; c < NCHUNK; ++c) {
        float ov = pval[base + (size_t)c * npts];
        int   oi = pidx[base + (size_t)c * npts];
        bool take = ov < v;                      // chunks ascend in index: strict < keeps first
        v  = take ? ov : v;
        ix = take ? oi : ix;
    }

    const int ntotal = NB * npts;
    float* dist = out + (size_t)dir * ntotal + rem;                    // dist1 | dist2
    int*   idx  = (int*)out + 2 * ntotal + (size_t)dir * ntotal + rem; // idx1 | idx2
    *dist = sqrtf(fmaxf(v, 0.0f));
    *idx  = ix;
}

extern "C" void kernel_launch(void* const* d_in, const int* in_sizes, int n_in,
                              void* d_out, int out_size, void* d_ws, size_t ws_size,
                              hipStream_t stream) {
    const float* xyz1 = (const float*)d_in[0];   // [4,8192,3] f32
    const float* xyz2 = (const float*)d_in[1];   // [4,8192,3] f32

    const int npts   = NPTS;
    const int ntotal = NB * NPTS;                // 32768 points per set

    // Workspace: pkA1 | pkB1 | pkA2 | pkB2 (float4 each) | pval | pidx
    float4* pkA1 = (float4*)d_ws;
    float4* pkB1 = pkA1 + ntotal;
    float4* pkA2 = pkB1 + ntotal;
    float4* pkB2 = pkA2 + ntotal;
    float*  pval = (float*)(pkB2 + ntotal);
    int*    pidx = (int*)(pval + (size_t)2 * NB * NCHUNK * npts);
    size_t need = (size_t)4 * ntotal * sizeof(float4)
                + (size_t)2 * 2 * NB * NCHUNK * npts * sizeof(float);
    if (ws_size < need) return;

    // Pack both operand forms + norms.
    {
        dim3 blk(256), grd((ntotal + 255) / 256);
        chamfer_pack_kernel<<<grd, blk, 0, stream>>>(xyz1, pkA1, pkB1, ntotal);
        chamfer_pack_kernel<<<grd, blk, 0, stream>>>(xyz2, pkA2, pkB2, ntotal);
    }

    // Fused directional/chunked partial pass: 32 x 4 x 8 workgroups of 8 waves.
    {
        dim3 blk(THREADS);
        dim3 grd(npts / (WAVES_PER_BLOCK * Q_PER_WAVE), NB, 2 * NCHUNK);
        chamfer_partial_kernel<<<grd, blk, 0, stream>>>(
            pkA1, pkB1, pkA2, pkB2, pval, pidx, npts);
    }

    // Final combine + clamp + sqrt.
    {
        int total = 2 * ntotal;
        dim3 blk(256), grd((total + 255) / 256);
        chamfer_reduce_kernel<<<grd, blk, 0, stream>>>(pval, pidx, (float*)d_out, npts);
    }
}